// GConv_81131932221715
// MI455X (gfx1250) — compile-verified
//
#include <hip/hip_runtime.h>

typedef __attribute__((ext_vector_type(2))) float v2f;
typedef __attribute__((ext_vector_type(8))) float v8f;

#define NODE_DIM 128

// ---------------------------------------------------------------------------
// Kernel A: SpMM gather. One wave32 per node. edge_rows is sorted, so each
// wave binary-searches its edge range [lower_bound(i), lower_bound(i+1)).
// Each lane owns 4 of the 128 features (float4 -> global_load_b128), so every
// edge gather is a single fully-coalesced 512B row read that lives in L2
// (h = 51.2MB < 192MB L2). No atomics, no workspace.
// ---------------------------------------------------------------------------
__global__ __launch_bounds__(256) void spmm_gather_kernel(
    const int* __restrict__ edge_rows, const int* __restrict__ edge_cols,
    const float* __restrict__ edge_vals, const float* __restrict__ h,
    float* __restrict__ agg, int n_edges, int n_nodes) {
  const int gtid = blockIdx.x * blockDim.x + threadIdx.x;
  const int node = gtid >> 5;           // wave32: one node per wave
  const int lane = threadIdx.x & 31;
  if (node >= n_nodes) return;

  // lower_bound(edge_rows, node)
  int lo = 0, hi = n_edges;
  while (lo < hi) {
    int mid = (lo + hi) >> 1;
    if (edge_rows[mid] < node) lo = mid + 1; else hi = mid;
  }
  const int e_begin = lo;
  // lower_bound(edge_rows, node + 1)
  hi = n_edges;
  while (lo < hi) {
    int mid = (lo + hi) >> 1;
    if (edge_rows[mid] <= node) lo = mid + 1; else hi = mid;
  }
  const int e_end = lo;

  float4 acc = make_float4(0.f, 0.f, 0.f, 0.f);
  const float4* __restrict__ h4 = (const float4*)h;
  for (int e = e_begin; e < e_end; ++e) {
    const float v = edge_vals[e];                       // uniform per wave
    const int   c = edge_cols[e];                       // uniform per wave
    const float4 hv = h4[(size_t)c * (NODE_DIM / 4) + lane];
    acc.x = fmaf(v, hv.x, acc.x);
    acc.y = fmaf(v, hv.y, acc.y);
    acc.z = fmaf(v, hv.z, acc.z);
    acc.w = fmaf(v, hv.w, acc.w);
  }
  ((float4*)agg)[(size_t)node * (NODE_DIM / 4) + lane] = acc;
}

// ---------------------------------------------------------------------------
// Kernel B: out = relu(agg @ W^T + b), IN PLACE on the agg buffer.
// 256 threads = 8 wave32 per block; block owns a 16-row tile of agg (staged
// in LDS behind a barrier, so the in-place overwrite is safe), each wave owns
// one 16x16 output tile (8 waves x 16 cols = 128 cols).
// K-loop: 32 x V_WMMA_F32_16X16X4_F32 with the C accumulator chained.
//
// Layouts per ISA 7.12.2 (f32, wave32):
//   A 16x4 : lane m = lane&15 ; half-wave picks K pair -> contiguous float2
//   B 4x16 : lane n = lane&15 ; half-wave picks K pair -> B[k][n] = W[n][k],
//            contiguous float2 from W's row n
//   C 16x16: VGPR i holds row i (lanes 0-15) / row i+8 (lanes 16-31)
// ---------------------------------------------------------------------------
__global__ __launch_bounds__(256) void gemm_bias_relu_kernel(
    float* __restrict__ inout, const float* __restrict__ W,
    const float* __restrict__ bias, int n_nodes) {
  // 132-float row pitch: (m*132 + k) % 64 banks varies with m -> conflict-free
  __shared__ __align__(16) float atile[16][132];

  const int rowBase = blockIdx.x * 16;
  if (rowBase >= n_nodes) return;

  // Cooperative stage of the 16x128 A tile (512 float4 over 256 threads).
  for (int i = threadIdx.x; i < 16 * 32; i += 256) {
    const int r  = i >> 5;
    const int c4 = i & 31;
    float4 v = make_float4(0.f, 0.f, 0.f, 0.f);
    if (rowBase + r < n_nodes)
      v = ((const float4*)inout)[(size_t)(rowBase + r) * (NODE_DIM / 4) + c4];
    *(float4*)&atile[r][c4 * 4] = v;
  }
  __syncthreads();

  const int lane  = threadIdx.x & 31;
  const int m16   = lane & 15;
  const int khalf = (lane >> 4) << 1;                 // 0 or 2
  const int nCol  = ((threadIdx.x >> 5) << 4) + m16;  // this wave's 16-col tile
  const float* __restrict__ wrow = W + (size_t)nCol * NODE_DIM;

  v8f c = {0.f, 0.f, 0.f, 0.f, 0.f, 0.f, 0.f, 0.f};
#pragma unroll
  for (int k0 = 0; k0 < NODE_DIM; k0 += 4) {
    v2f a = *(const v2f*)&atile[m16][k0 + khalf];     // ds_load_b64
    v2f b = *(const v2f*)&wrow[k0 + khalf];           // global_load_b64 (L2-resident)
    c = __builtin_amdgcn_wmma_f32_16x16x4_f32(
        /*neg_a=*/false, a, /*neg_b=*/false, b,
        /*c_mod=*/(short)0, c, /*reuse_a=*/false, /*reuse_b=*/false);
  }

  const float bv = bias[nCol];
  const int rowOff = (lane >> 4) << 3;                // 0 or 8
#pragma unroll
  for (int i = 0; i < 8; ++i) {
    const int row = rowBase + rowOff + i;
    float o = c[i] + bv;
    o = fmaxf(o, 0.f);
    if (row < n_nodes) inout[(size_t)row * NODE_DIM + nCol] = o;
  }
}

// ---------------------------------------------------------------------------
extern "C" void kernel_launch(void* const* d_in, const int* in_sizes, int n_in,
                              void* d_out, int out_size, void* d_ws, size_t ws_size,
                              hipStream_t stream) {
  const int*   edge_rows = (const int*)d_in[0];
  const int*   edge_cols = (const int*)d_in[1];
  const float* edge_vals = (const float*)d_in[2];
  const float* h         = (const float*)d_in[3];
  const float* W         = (const float*)d_in[4];
  const float* bias      = (const float*)d_in[5];
  float*       out       = (float*)d_out;

  const int n_edges = in_sizes[0];
  const int n_nodes = in_sizes[3] / NODE_DIM;

  // SpMM: one wave per node, 8 waves (256 threads) per block.
  {
    const int threads = 256;
    const int blocks  = (n_nodes * 32 + threads - 1) / threads;
    spmm_gather_kernel<<<blocks, threads, 0, stream>>>(
        edge_rows, edge_cols, edge_vals, h, out, n_edges, n_nodes);
  }
  // Fused GEMM + bias + ReLU, in place on d_out.
  {
    const int blocks = (n_nodes + 15) / 16;
    gemm_bias_relu_kernel<<<blocks, 256, 0, stream>>>(out, W, bias, n_nodes);
  }
}